// DCN_63410897158364
// MI455X (gfx1250) — compile-verified
//
#include <hip/hip_runtime.h>

// DCNv2 forward, fused for MI455X (gfx1250, wave32).
// Sizes fixed by the reference:
#define B_    4
#define CIN_  256
#define COUT_ 256
#define H_    64
#define W_    64
#define HO_   64
#define WO_   64
#define K2_   9
#define KTOT  (CIN_*K2_)          // 2304

typedef __attribute__((ext_vector_type(2))) float v2f;
typedef __attribute__((ext_vector_type(8))) float v8f;

// ---------------------------------------------------------------------------
// Kernel A: 3x3 offset conv (27 channels) -> dy / dx / sigmoid(mask).
// offset = om[:,0:18]; dy[tap]=om[2*tap], dx[tap]=om[2*tap+1], mask=sig(om[18+t]).
// grid (ho=64, oc=27, b=4), block 64 (wo). Weight reads are wave-uniform.
// ---------------------------------------------------------------------------
__global__ __launch_bounds__(64) void dcn_offset_conv(
    const float* __restrict__ x, const float* __restrict__ w,
    const float* __restrict__ bias,
    float* __restrict__ dY, float* __restrict__ dX, float* __restrict__ MK)
{
  const int wo = threadIdx.x;
  const int ho = blockIdx.x;
  const int oc = blockIdx.y;
  const int b  = blockIdx.z;

  float acc = bias[oc];
  const float* wp0 = w + (size_t)oc * CIN_ * K2_;
  const float* xp0 = x + (size_t)b  * CIN_ * H_ * W_;
  for (int c = 0; c < CIN_; ++c) {
    const float* wp = wp0 + c * 9;
    const float* xp = xp0 + c * H_ * W_;
#pragma unroll
    for (int ki = 0; ki < 3; ++ki) {
      const int y = ho - 1 + ki;
      if ((unsigned)y < (unsigned)H_) {
#pragma unroll
        for (int kj = 0; kj < 3; ++kj) {
          const int xx = wo - 1 + kj;
          if ((unsigned)xx < (unsigned)W_)
            acc = fmaf(wp[ki * 3 + kj], xp[y * W_ + xx], acc);
        }
      }
    }
  }
  const int pix = (ho << 6) + wo;
  if (oc < 18) {
    const int tap = oc >> 1;
    float* dst = (oc & 1) ? dX : dY;
    dst[((b * 9 + tap) << 12) + pix] = acc;
  } else {
    const int tap = oc - 18;
    MK[((b * 9 + tap) << 12) + pix] = 1.0f / (1.0f + __expf(-acc));
  }
}

// ---------------------------------------------------------------------------
// Kernel B: fused bilinear-sampling + GEMM with fp32 WMMA (16x16x4).
// One workgroup = one (b, ho) output row: N = 64 pixels, M = 256 COUT.
// 8 waves in a 4(M) x 2(N) grid; each wave: M=64 x N=32 -> 8 v8f accumulators.
// K = 2304 walked in chunks of 36 = 4 channels * 9 taps (k = c*9 + tap,
// matching w_dcn.reshape(COUT, CIN*K2)).
// ---------------------------------------------------------------------------
#define KC 36
#define CSTR 72   // cols LDS row stride: 2*72 = 144 = 16 (mod 64) -> lanes 0-15
                  // (row k) and 16-31 (row k+2) hit disjoint LDS banks.

__global__ __launch_bounds__(256) void dcn_main(
    const float* __restrict__ x, const float* __restrict__ wdcn,
    const float* __restrict__ bdcn,
    const float* __restrict__ dY, const float* __restrict__ dX,
    const float* __restrict__ MK, float* __restrict__ out)
{
  __shared__ int   s_idx[K2_ * WO_][4];   // clamped corner indices (y*W+x)
  __shared__ float s_w  [K2_ * WO_][4];   // bilinear weights * mask (0 if OOB)
  __shared__ float s_cols[KC * CSTR];     // cols chunk [36][64] (padded)

  const int tid = threadIdx.x;
  const int ho  = blockIdx.x;
  const int b   = blockIdx.y;

  // ---- phase 1: bilinear meta per (tap, n) ----
  for (int e = tid; e < K2_ * WO_; e += 256) {
    const int tap = e >> 6;
    const int n   = e & 63;
    const int base = ((b * 9 + tap) << 12) + (ho << 6) + n;
    const float mv  = MK[base];
    const float py  = dY[base] + (float)(ho - 1 + tap / 3);
    const float px  = dX[base] + (float)(n  - 1 + tap % 3);
    const float y0f = floorf(py), x0f = floorf(px);
    const int y0 = (int)y0f, x0 = (int)x0f;
    const int y1 = y0 + 1,   x1 = x0 + 1;
    const float ly = py - y0f, lx = px - x0f;
    const float hy = 1.0f - ly, hx = 1.0f - lx;
    const bool vy0 = (unsigned)y0 < (unsigned)H_, vy1 = (unsigned)y1 < (unsigned)H_;
    const bool vx0 = (unsigned)x0 < (unsigned)W_, vx1 = (unsigned)x1 < (unsigned)W_;
    const int cy0 = min(max(y0, 0), H_ - 1), cy1 = min(max(y1, 0), H_ - 1);
    const int cx0 = min(max(x0, 0), W_ - 1), cx1 = min(max(x1, 0), W_ - 1);
    s_idx[e][0] = cy0 * W_ + cx0;  s_idx[e][1] = cy0 * W_ + cx1;
    s_idx[e][2] = cy1 * W_ + cx0;  s_idx[e][3] = cy1 * W_ + cx1;
    s_w[e][0] = (vy0 && vx0) ? hy * hx * mv : 0.0f;
    s_w[e][1] = (vy0 && vx1) ? hy * lx * mv : 0.0f;
    s_w[e][2] = (vy1 && vx0) ? ly * hx * mv : 0.0f;
    s_w[e][3] = (vy1 && vx1) ? ly * lx * mv : 0.0f;
  }
  __syncthreads();

  const int lane  = tid & 31;
  const int wid   = tid >> 5;
  const int wm    = wid & 3;          // M wave coord (0..3) -> 64 rows each
  const int wn    = wid >> 2;         // N wave coord (0..1) -> 32 cols each
  const int mhalf = lane >> 4;        // 0: K=0/1 half, 1: K=2/3 half
  const int l16   = lane & 15;

  v8f acc[4][2];
#pragma unroll
  for (int i = 0; i < 4; ++i)
#pragma unroll
    for (int j = 0; j < 2; ++j)
#pragma unroll
      for (int q = 0; q < 8; ++q) acc[i][j][q] = 0.0f;

  const int nB = tid & 63;            // pixel this thread fills in cols
  const int kg = tid >> 6;            // 0..3 -> channel c0+kg of the chunk
  const float* xb = x + (size_t)b * CIN_ * H_ * W_;

  for (int c0 = 0; c0 < CIN_; c0 += 4) {
    // ---- build cols chunk: k' = (ci-c0)*9 + tap, n = pixel ----
    {
      const float* xc = xb + ((size_t)(c0 + kg) << 12);
#pragma unroll
      for (int tap = 0; tap < 9; ++tap) {
        const int e  = (tap << 6) + nB;
        const float v = s_w[e][0] * xc[s_idx[e][0]]
                      + s_w[e][1] * xc[s_idx[e][1]]
                      + s_w[e][2] * xc[s_idx[e][2]]
                      + s_w[e][3] * xc[s_idx[e][3]];
        s_cols[(kg * 9 + tap) * CSTR + nB] = v;
      }
    }
    __syncthreads();

    // ---- 9 WMMA K-steps of 4 over this chunk ----
#pragma unroll
    for (int ks = 0; ks < 9; ++ks) {
      const int k4    = ks * 4;
      const int kglob = c0 * 9 + k4 + mhalf * 2;   // even -> 8B aligned
      // A fragments: 16x4 fp32 layout. lane<16: M=l16,K={0,1}; lane>=16: K={2,3}.
      v2f afr[4];
#pragma unroll
      for (int mt = 0; mt < 4; ++mt) {
        const int m = wm * 64 + mt * 16 + l16;
        afr[mt] = *(const v2f*)(wdcn + (size_t)m * KTOT + kglob);
      }
#pragma unroll
      for (int nt = 0; nt < 2; ++nt) {
        // B fragment: 4x16 fp32. lane<16: N=l16,K={0,1}; lane>=16: K={2,3}.
        const int n = wn * 32 + nt * 16 + l16;
        v2f bfr;
        bfr.x = s_cols[(k4 + mhalf * 2    ) * CSTR + n];
        bfr.y = s_cols[(k4 + mhalf * 2 + 1) * CSTR + n];
#pragma unroll
        for (int mt = 0; mt < 4; ++mt) {
          acc[mt][nt] = __builtin_amdgcn_wmma_f32_16x16x4_f32(
              false, afr[mt], false, bfr, (short)0, acc[mt][nt], false, false);
        }
      }
    }
    __syncthreads();   // cols chunk fully consumed before rebuild
  }

  // ---- epilogue: C/D layout: VGPR v, lanes>=16 add M+8; N = l16 ----
#pragma unroll
  for (int mt = 0; mt < 4; ++mt) {
#pragma unroll
    for (int nt = 0; nt < 2; ++nt) {
      const int n = wn * 32 + nt * 16 + l16;
#pragma unroll
      for (int v = 0; v < 8; ++v) {
        const int m = wm * 64 + mt * 16 + mhalf * 8 + v;
        out[(((size_t)(b * COUT_ + m)) * HO_ + ho) * WO_ + n] =
            acc[mt][nt][v] + bdcn[m];
      }
    }
  }
}

// ---------------------------------------------------------------------------
extern "C" void kernel_launch(void* const* d_in, const int* in_sizes, int n_in,
                              void* d_out, int out_size, void* d_ws, size_t ws_size,
                              hipStream_t stream) {
  const float* x    = (const float*)d_in[0];   // (4,256,64,64)
  const float* woff = (const float*)d_in[1];   // (27,256,3,3)
  const float* boff = (const float*)d_in[2];   // (27,)
  const float* wdcn = (const float*)d_in[3];   // (256,256,3,3)
  const float* bdcn = (const float*)d_in[4];   // (256,)
  float* out = (float*)d_out;                  // (4,256,64,64)

  // workspace: dy / dx / mask, each B*9*HO*WO floats (1.77 MB total)
  float* ws = (float*)d_ws;
  const size_t plane = (size_t)B_ * 9 * HO_ * WO_;
  float* dY = ws;
  float* dX = ws + plane;
  float* MK = ws + 2 * plane;

  dim3 gA(HO_, 27, B_);
  hipLaunchKernelGGL(dcn_offset_conv, gA, dim3(64), 0, stream,
                     x, woff, boff, dY, dX, MK);

  dim3 gB(HO_, B_);
  hipLaunchKernelGGL(dcn_main, gB, dim3(256), 0, stream,
                     x, wdcn, bdcn, dY, dX, MK, out);
}